// GraphMatcher_25718264169334
// MI455X (gfx1250) — compile-verified
//
#include <hip/hip_runtime.h>
#include <math.h>

#define N   32
#define NN  1024          // n*n
#define MAX_ITER 20
#define WAVES_A 4         // waves per block in edge kernel

// ---------------------------------------------------------------------------
// CDNA5 async-to-LDS helpers (gfx1250). Each instruction copies 16B per lane
// (512B per wave32). Tracked by ASYNCcnt; loads complete in order.
// ---------------------------------------------------------------------------
__device__ __forceinline__ void async_ld16(unsigned lds_off, const void* g) {
  asm volatile("global_load_async_to_lds_b128 %0, %1, off"
               :: "v"(lds_off), "v"((unsigned long long)(size_t)g)
               : "memory");
}

template <int CNT>
__device__ __forceinline__ void async_wait() {
  asm volatile("s_wait_asynccnt %0" :: "n"(CNT) : "memory");
}

// ---------------------------------------------------------------------------
// Edge kernel: one edge per wave per step.
//   msgs[e][i] = max_j x[dst[e]][j] * K[e][i*32+j]      (lane i = row i)
//   acc[src[e]*32 + i] += msgs[e][i]                    (global f32 atomic)
// K block (4KB/edge) is streamed through LDS with async loads, double
// buffered per wave. Rows are stored padded to 144B in LDS so the per-lane
// ds_load_b128 row reads (stride 144B) avoid 16-way bank conflicts.
// ---------------------------------------------------------------------------
__global__ __launch_bounds__(WAVES_A * 32) void mpm_edges(
    const float* __restrict__ K, const int* __restrict__ ei,
    const float* __restrict__ x, float* acc, int E)
{
  // 4 waves * 2 slots * 32 rows * 144B = 36864 B
  __shared__ __align__(16) unsigned char kls[WAVES_A][2][N * 144];

  const int lane = threadIdx.x & 31;
  const int wid  = threadIdx.x >> 5;
  const int gw   = blockIdx.x * WAVES_A + wid;   // global wave id
  const int TOT  = gridDim.x * WAVES_A;          // total waves

  // async-copy geometry: chunk k, lane l covers float index k*128 + l*4
  //   -> row = k*4 + (l>>3), byte-col = (l&7)*16, LDS row stride 144B
  const unsigned rsub = (unsigned)(lane >> 3);
  const unsigned colb = (unsigned)(lane & 7) * 16u;
  unsigned base[2];
  base[0] = (unsigned)(size_t)&kls[wid][0][0];
  base[1] = (unsigned)(size_t)&kls[wid][1][0];

  // prefetch first edge into slot 0
  if (gw < E) {
    const char* gp = (const char*)K + (size_t)gw * 4096 + (size_t)lane * 16;
#pragma unroll
    for (int k = 0; k < 8; ++k)
      async_ld16(base[0] + ((unsigned)k * 4u + rsub) * 144u + colb, gp + k * 512);
  }

  int slot = 0;
  for (int e = gw; e < E; e += TOT, slot ^= 1) {
    const int en = e + TOT;
    if (en < E) {   // prefetch next edge into the other slot
      const char* gp = (const char*)K + (size_t)en * 4096 + (size_t)lane * 16;
#pragma unroll
      for (int k = 0; k < 8; ++k)
        async_ld16(base[slot ^ 1] + ((unsigned)k * 4u + rsub) * 144u + colb,
                   gp + k * 512);
    }

    // edge endpoints are wave-uniform -> force into SGPRs
    const int srcN = __builtin_amdgcn_readfirstlane(ei[e]);
    const int dstN = __builtin_amdgcn_readfirstlane(ei[E + e]);

    // x[dst] row: SGPR base + per-lane (uniform) offsets, broadcast loads
    const float4* xr = (const float4*)(x + (size_t)dstN * N);
    float4 xd[8];
#pragma unroll
    for (int k = 0; k < 8; ++k) xd[k] = xr[k];

    // current edge's 8 async chunks are the oldest 8 outstanding
    if (en < E) async_wait<8>(); else async_wait<0>();

    const float4* kr = (const float4*)&kls[wid][slot][lane * 144];

    // max-tree with short dependency chains (fmax is associative; the
    // compiler won't reassociate FP max, so do it explicitly for ILP)
    float pk[8];
#pragma unroll
    for (int k = 0; k < 8; ++k) {
      const float4 kv = kr[k];
      const float a = kv.x * xd[k].x;
      const float b = kv.y * xd[k].y;
      const float c = kv.z * xd[k].z;
      const float d = kv.w * xd[k].w;
      pk[k] = fmaxf(fmaxf(a, b), fmaxf(c, d));
    }
    const float m01 = fmaxf(pk[0], pk[1]);
    const float m23 = fmaxf(pk[2], pk[3]);
    const float m45 = fmaxf(pk[4], pk[5]);
    const float m67 = fmaxf(pk[6], pk[7]);
    const float m   = fmaxf(fmaxf(m01, m23), fmaxf(m45, m67));

    atomicAdd(acc + (size_t)srcN * N + lane, m);
  }
}

// ---------------------------------------------------------------------------
// Per-graph L2 normalization: xo[b] = src[b] / ||src[b]||_2, and zero acc.
// One block per graph, 256 threads * float4 = 1024 elements.
// (src may alias acc — read before zeroing.)
// ---------------------------------------------------------------------------
__global__ __launch_bounds__(256) void normalize_k(
    const float* src, float* __restrict__ xo, float* acc)
{
  __shared__ float wsum[8];
  const int b = blockIdx.x, t = threadIdx.x;

  float4 v = ((const float4*)(src + (size_t)b * NN))[t];
  float ss = v.x * v.x + v.y * v.y + v.z * v.z + v.w * v.w;
#pragma unroll
  for (int m = 16; m; m >>= 1) ss += __shfl_xor(ss, m, 32);
  if ((t & 31) == 0) wsum[t >> 5] = ss;
  __syncthreads();

  float tot = 0.0f;
#pragma unroll
  for (int k = 0; k < 8; ++k) tot += wsum[k];
  const float rn = rsqrtf(tot);

  float4 o; o.x = v.x * rn; o.y = v.y * rn; o.z = v.z * rn; o.w = v.w * rn;
  ((float4*)(xo + (size_t)b * NN))[t] = o;

  float4 z; z.x = 0.f; z.y = 0.f; z.z = 0.f; z.w = 0.f;
  ((float4*)(acc + (size_t)b * NN))[t] = z;
}

// ---------------------------------------------------------------------------
// Output: out[b, j, i] = x[b*32 + i][j]  (per-graph transpose through LDS)
// ---------------------------------------------------------------------------
__global__ __launch_bounds__(256) void transpose_out(
    const float* __restrict__ x, float* __restrict__ out)
{
  __shared__ float tile[NN];
  const int b = blockIdx.x, t = threadIdx.x;
  ((float4*)tile)[t] = ((const float4*)(x + (size_t)b * NN))[t];
  __syncthreads();
#pragma unroll
  for (int k = 0; k < 4; ++k) {
    const int o = t * 4 + k;                    // o = j*32 + i
    out[(size_t)b * NN + o] = tile[(o & 31) * 32 + (o >> 5)];
  }
}

// ---------------------------------------------------------------------------
extern "C" void kernel_launch(void* const* d_in, const int* in_sizes, int n_in,
                              void* d_out, int out_size, void* d_ws, size_t ws_size,
                              hipStream_t stream)
{
  const float* K  = (const float*)d_in[0];
  const int*   ei = (const int*)d_in[1];     // (2, E) int32
  const float* x0 = (const float*)d_in[2];   // (bsz*n, n)
  float* out = (float*)d_out;

  const int E   = in_sizes[1] / 2;
  const int BN  = in_sizes[2];               // bsz * n * n
  const int bsz = BN / NN;

  float* acc = (float*)d_ws;                 // BN floats
  float* xw  = acc + BN;                     // BN floats

  // init: normalize x0 per graph, zero accumulator
  normalize_k<<<bsz, 256, 0, stream>>>(x0, xw, acc);

  // 2304 blocks * 4 waves = 9216 waves -> 4 edges per wave over E=36864,
  // so the async double-buffer pipeline stays primed for 4 steps.
  const int gridA = 2304;
  for (int it = 0; it < MAX_ITER; ++it) {
    mpm_edges<<<gridA, WAVES_A * 32, 0, stream>>>(K, ei, xw, acc, E);
    normalize_k<<<bsz, 256, 0, stream>>>(acc, xw, acc);
  }

  transpose_out<<<bsz, 256, 0, stream>>>(xw, out);
}